// AttentionLayer_36361193128336
// MI455X (gfx1250) — compile-verified
//
#include <hip/hip_runtime.h>
#include <hip/hip_bf16.h>

typedef __attribute__((ext_vector_type(2))) float v2f;
typedef __attribute__((ext_vector_type(4))) float v4f;
typedef __attribute__((ext_vector_type(8))) float v8f;

#define NB 8
#define SS 2048
#define HH 1024

#define BM 128
#define BN 128
#define BK 16
#define ASTR 20   /* floats per A row: 16 + pad4; 80B rows stay 16B aligned,   */
                  /* and 20*dm = 2,3 (mod 64) has no solution -> no conflicts  */
#define BSTR 144  /* float2 per B k-pair row: 2*144 = 32 (mod 64) -> the two   */
                  /* half-wave K rows hit disjoint bank halves                  */
#define ASZ (BM * ASTR)      /* floats per A buffer  */
#define BSZ (8 * BSTR * 2)   /* floats per B buffer (8 k-pairs x 144 float2)   */

__device__ __forceinline__ v8f wmma4(v2f a, v2f b, v8f c) {
  // V_WMMA_F32_16X16X4_F32: D = A(16x4,f32) * B(4x16,f32) + C(16x16,f32)
  return __builtin_amdgcn_wmma_f32_16x16x4_f32(
      false, a, false, b, (short)0, c, false, false);
}

// C[m,n] = sum_k A[m,k] * B[k,n]
//   BT==0 : B[k][n] = Bm[k*ldb + n]   (NN)
//   BT==1 : B[k][n] = Bm[n*ldb + k]   (NT: B supplied row-major [n][k])
template <int BT>
__global__ __launch_bounds__(256)
void gemm_f32_wmma(const float* __restrict__ A, int lda, size_t sA,
                   const float* __restrict__ Bm, int ldb, size_t sB,
                   float* __restrict__ C, int ldc, size_t sC, int K) {
  __shared__ float As[2 * ASZ];
  __shared__ float Bs[2 * BSZ];

  const int tid  = threadIdx.x;
  const int lane = tid & 31;
  const int wid  = tid >> 5;
  const int wm   = (wid & 3) * 32;   // wave M offset in 128x128 tile
  const int wn   = (wid >> 2) * 64;  // wave N offset
  const int lr   = lane & 15;        // row/col within a 16-wide fragment
  const int hi   = lane >> 4;        // 0: K{0,1}  1: K{2,3}

  const int    bn = blockIdx.x * BN;
  const float* Ab = A + blockIdx.z * sA + (size_t)(blockIdx.y * BM) * lda;
  const float* Bb = Bm + blockIdx.z * sB;
  float*       Cb = C + blockIdx.z * sC + (size_t)(blockIdx.y * BM) * ldc + bn;

  // --- staging thread mappings ---
  const int arow = tid >> 2;          // A: rows arow, arow+64
  const int ac4  = (tid & 3) * 4;     // A: 4 consecutive k
  const int bnt  = tid & 127;         // B/NT: one x-row (= one output col)
  const int bkq  = (tid >> 7) * 8;    // B/NT: 8 consecutive k
  const int bkr  = tid >> 4;          // B/NN: one k row
  const int bc8  = (tid & 15) * 8;    // B/NN: 8 consecutive n

  v4f ar0, ar1, br0, br1;

  auto gload = [&](int k0) {
    ar0 = *(const v4f*)(Ab + (size_t)arow * lda + (k0 + ac4));
    ar1 = *(const v4f*)(Ab + (size_t)(arow + 64) * lda + (k0 + ac4));
    if (BT) {
      const float* p = Bb + (size_t)(bn + bnt) * ldb + (k0 + bkq);
      br0 = *(const v4f*)(p);
      br1 = *(const v4f*)(p + 4);
    } else {
      const float* p = Bb + (size_t)(k0 + bkr) * ldb + (bn + bc8);
      br0 = *(const v4f*)(p);
      br1 = *(const v4f*)(p + 4);
    }
  };

  auto lstore = [&](float* Ad, float* Bd) {
    *(v4f*)(Ad + arow * ASTR + ac4) = ar0;           // 16B-aligned b128 store
    *(v4f*)(Ad + (arow + 64) * ASTR + ac4) = ar1;
    if (BT) {
      // column bnt gets k-pairs (bkq/2 .. bkq/2+3), each a contiguous float2
      float* q = Bd + ((bkq >> 1) * BSTR + bnt) * 2;
      v2f p0; p0.x = br0.x; p0.y = br0.y;
      v2f p1; p1.x = br0.z; p1.y = br0.w;
      v2f p2; p2.x = br1.x; p2.y = br1.y;
      v2f p3; p3.x = br1.z; p3.y = br1.w;
      *(v2f*)(q + 0 * BSTR * 2) = p0;
      *(v2f*)(q + 1 * BSTR * 2) = p1;
      *(v2f*)(q + 2 * BSTR * 2) = p2;
      *(v2f*)(q + 3 * BSTR * 2) = p3;
    } else {
      // row bkr contributes component (bkr&1) of pairs (bkr>>1, n)
      float* q = Bd + (bkr >> 1) * BSTR * 2 + (bkr & 1);
      q[(bc8 + 0) * 2] = br0.x;  q[(bc8 + 1) * 2] = br0.y;
      q[(bc8 + 2) * 2] = br0.z;  q[(bc8 + 3) * 2] = br0.w;
      q[(bc8 + 4) * 2] = br1.x;  q[(bc8 + 5) * 2] = br1.y;
      q[(bc8 + 6) * 2] = br1.z;  q[(bc8 + 7) * 2] = br1.w;
    }
  };

  v8f acc[2][4] = {};

  gload(0);
  lstore(As, Bs);
  __syncthreads();

  const int nk = K >> 4;
  for (int kt = 0; kt < nk; ++kt) {
    const int    cur = kt & 1;
    const float* Ac  = As + cur * ASZ + (wm + lr) * ASTR;     // + 16*ASTR for mi=1
    const float* Bc  = Bs + cur * BSZ + (hi * BSTR + wn + lr) * 2;

    if (kt + 1 < nk) gload((kt + 1) << 4);   // overlap VMEM with WMMA

#pragma unroll
    for (int k4 = 0; k4 < BK; k4 += 4) {
      // A fragment: lane row = wm(+16)+lr, K pair = k4 + 2*hi  (8B ds_load)
      v2f a0 = *(const v2f*)(Ac + k4 + 2 * hi);
      v2f a1 = *(const v2f*)(Ac + 16 * ASTR + k4 + 2 * hi);
      const float* Bk = Bc + (k4 >> 1) * BSTR * 2;
#pragma unroll
      for (int ni = 0; ni < 4; ++ni) {
        v2f b = *(const v2f*)(Bk + ni * 32);  // pair-interleaved: one b64 load
        acc[0][ni] = wmma4(a0, b, acc[0][ni]);
        acc[1][ni] = wmma4(a1, b, acc[1][ni]);
      }
    }
    __syncthreads();
    if (kt + 1 < nk) {
      lstore(As + (cur ^ 1) * ASZ, Bs + (cur ^ 1) * BSZ);
      __syncthreads();
    }
  }

  // writeback: wave owns 32x64; vgpr r -> row r (+8 for upper half-wave)
#pragma unroll
  for (int mi = 0; mi < 2; ++mi) {
#pragma unroll
    for (int ni = 0; ni < 4; ++ni) {
      const int n = wn + ni * 16 + lr;
#pragma unroll
      for (int r = 0; r < 8; ++r) {
        const int m = wm + mi * 16 + hi * 8 + r;
        Cb[(size_t)m * ldc + n] = acc[mi][ni][r];
      }
    }
  }
}

// In-place row softmax over rows of length 2048. One block (256 thr) per row.
__global__ __launch_bounds__(256)
void softmax_rows_2048(float* __restrict__ attn) {
  float* p = attn + (size_t)blockIdx.x * SS;
  const int tid  = threadIdx.x;
  const int lane = tid & 31;
  const int wid  = tid >> 5;
  __shared__ float red[8];

  float v[8];
  float mx = -3.402823466e+38f;
#pragma unroll
  for (int i = 0; i < 8; ++i) {
    v[i] = p[tid + i * 256];
    mx = fmaxf(mx, v[i]);
  }
#pragma unroll
  for (int off = 16; off >= 1; off >>= 1) mx = fmaxf(mx, __shfl_xor(mx, off, 32));
  if (lane == 0) red[wid] = mx;
  __syncthreads();
  mx = red[0];
#pragma unroll
  for (int i = 1; i < 8; ++i) mx = fmaxf(mx, red[i]);
  __syncthreads();

  float s = 0.f;
#pragma unroll
  for (int i = 0; i < 8; ++i) {
    v[i] = __expf(v[i] - mx);
    s += v[i];
  }
#pragma unroll
  for (int off = 16; off >= 1; off >>= 1) s += __shfl_xor(s, off, 32);
  if (lane == 0) red[wid] = s;
  __syncthreads();
  s = red[0];
#pragma unroll
  for (int i = 1; i < 8; ++i) s += red[i];

  const float inv = 1.0f / s;
#pragma unroll
  for (int i = 0; i < 8; ++i) p[tid + i * 256] = v[i] * inv;
}

extern "C" void kernel_launch(void* const* d_in, const int* in_sizes, int n_in,
                              void* d_out, int out_size, void* d_ws, size_t ws_size,
                              hipStream_t stream) {
  (void)in_sizes; (void)n_in; (void)out_size; (void)ws_size;
  const float* x = (const float*)d_in[0];   // [8,2048,1024]
  const float* W = (const float*)d_in[1];   // [1024,1024]
  float* ctx  = (float*)d_out;                         // [8,2048,1024]
  float* attn = (float*)d_out + (size_t)NB * SS * HH;  // [8,2048,2048]
  float* xw   = (float*)d_ws;                          // [8,2048,1024] scratch

  const size_t sXH = (size_t)SS * HH;
  const size_t sAT = (size_t)SS * SS;

  dim3 blk(256);

  // 1) xw[b] = x[b] @ W              M=2048 N=1024 K=1024 (W broadcast: sB=0)
  gemm_f32_wmma<0><<<dim3(HH / BN, SS / BM, NB), blk, 0, stream>>>(
      x, HH, sXH, W, HH, (size_t)0, xw, HH, sXH, HH);

  // 2) scores[b] = xw[b] @ x[b]^T    M=2048 N=2048 K=1024 -> attn region
  gemm_f32_wmma<1><<<dim3(SS / BN, SS / BM, NB), blk, 0, stream>>>(
      xw, HH, sXH, x, HH, sXH, attn, SS, sAT, HH);

  // 3) softmax rows (in place)
  softmax_rows_2048<<<dim3(NB * SS), blk, 0, stream>>>(attn);

  // 4) context[b] = attn[b] @ x[b]   M=2048 N=1024 K=2048
  gemm_f32_wmma<0><<<dim3(HH / BN, SS / BM, NB), blk, 0, stream>>>(
      attn, SS, sAT, x, HH, sXH, ctx, HH, sXH, SS);
}